// Quanv_42159398978039
// MI455X (gfx1250) — compile-verified
//
#include <hip/hip_runtime.h>
#include <math.h>

#define NQ 9
#define NSTATE 512
#define ROWS_PER_BLOCK 8

// gfx1250 async global->LDS path (guarded: host pass / older toolchains fall back)
#if defined(__AMDGCN__) && __has_builtin(__builtin_amdgcn_global_load_async_to_lds_b32) && __has_builtin(__builtin_amdgcn_s_wait_asynccnt)
#define USE_ASYNC_LDS 1
#else
#define USE_ASYNC_LDS 0
#endif

typedef __attribute__((address_space(1))) int* gptr_i32;
typedef __attribute__((address_space(3))) int* lptr_i32;

// Weights-only tables, computed once per launch by quanv_prep.
__device__ float g_qtab[NQ * NSTATE];
__device__ int   g_utab[16];

// sigma_L: index map of one CNOT-ring layer (input-index as function of
// output-index). Qubit q lives at bit (8-q). Applies g_8 ... g_0.
__device__ __forceinline__ int sigma_layer(int j) {
#pragma unroll
  for (int i = 8; i >= 0; --i) {
    const int pc = 8 - i;
    const int pt = 8 - ((i + 1) % 9);
    j ^= ((j >> pc) & 1) << pt;
  }
  return j;
}

// Diagonal-phase angle of one RZ layer at basis index k.
__device__ __forceinline__ float phase_diag(const float* w9, int k) {
  float a = 0.f;
#pragma unroll
  for (int q = 0; q < 9; ++q)
    a += (((k >> (8 - q)) & 1) ? 0.5f : -0.5f) * w9[q];
  return a;
}

__global__ __launch_bounds__(512) void quanv_prep(const float* __restrict__ W) {
  __shared__ float As[NSTATE];
  __shared__ float Qs[NQ][NSTATE];
  __shared__ float w0[NQ], w1[NQ];
  const int j = threadIdx.x;
  if (j < NQ) { w0[j] = W[j]; w1[j] = W[NQ + j]; }
#pragma unroll
  for (int i = 0; i < NQ; ++i) Qs[i][j] = 0.f;
  __syncthreads();

  const int sL = sigma_layer(j);        // sigma_L(j)
  const int sj = sigma_layer(sL);       // sigma(j) = sigma_L^2(j)
  // final[j] = exp(i*A[j]) * m[sigma(j)];  (-i)^pop from RX state prep.
  As[j] = phase_diag(w1, sL) + phase_diag(w0, sj)
        - 1.5707963267948966f * (float)__popc(sj);
  if (j < NQ) g_utab[j] = sigma_layer(sigma_layer(1 << (8 - j)));
  __syncthreads();

#pragma unroll
  for (int i = 0; i < NQ; ++i) {
    const int bi = 1 << (8 - i);
    if ((j & bi) == 0)                 // 256 pair-terms per qubit, scattered to k-space
      Qs[i][sj] = 2.0f * sinf(As[j | bi] - As[j]);
  }
  __syncthreads();
#pragma unroll
  for (int i = 0; i < NQ; ++i) g_qtab[i * NSTATE + j] = Qs[i][j];
}

__global__ __launch_bounds__(256) void quanv_main(const float* __restrict__ x,
                                                  float* __restrict__ out) {
  __shared__ float Qs[NQ * NSTATE];    // 18 KB coefficient tables (broadcast reads)
  __shared__ float Ms[8 * NSTATE];     // 16 KB: per-wave real amplitude vector m[512]
  __shared__ int   Us[16];
  const int tid = threadIdx.x;

  // Stage Q tables into LDS; async on gfx1250 so it overlaps the trig below.
#if USE_ASYNC_LDS
#pragma unroll
  for (int it = 0; it < 18; ++it) {
    const int idx = it * 256 + tid;
    __builtin_amdgcn_global_load_async_to_lds_b32(
        (gptr_i32)&g_qtab[idx],
        (lptr_i32)&Qs[idx], 0, 0);
  }
#else
#pragma unroll
  for (int it = 0; it < 18; ++it) {
    const int idx = it * 256 + tid;
    Qs[idx] = g_qtab[idx];
  }
#endif
  if (tid < NQ) Us[tid] = g_utab[tid];

  const int wave = tid >> 5;
  const int lane = tid & 31;
  const int row  = blockIdx.x * ROWS_PER_BLOCK + wave;   // exact: 1568*8 = 12544
  const int b  = row / 784;
  const int hw = row - b * 784;
  const int h  = hw / 28;
  const int wc = hw - h * 28;

  // 3x3 patch angles (zero-padded): ONE sincos per lane (lanes 0..8 own one
  // patch element each), then wave32 broadcast — collapses 18 serial
  // transcendental latencies into 1 + shuffles.
  float c_own, s_own;
  {
    const int p  = (lane < NQ) ? lane : 0;
    const int hh = h + p / 3 - 1;
    const int ww = wc + p % 3 - 1;
    float ang = 0.f;
    if (hh >= 0 && hh < 28 && ww >= 0 && ww < 28) ang = x[b * 784 + hh * 28 + ww];
    sincosf(0.5f * ang, &s_own, &c_own);
  }
  float cq[NQ], sq[NQ];
#pragma unroll
  for (int p = 0; p < NQ; ++p) {
    cq[p] = __shfl(c_own, p, 32);
    sq[p] = __shfl(s_own, p, 32);
  }

  // m[k] = prod_q (bit_q(k) ? sin : cos).  k = t*32 + lane:
  // qubits 4..8 -> lane bits (shared factor), qubits 0..3 -> t bits (unrolled).
  float lf = 1.f;
#pragma unroll
  for (int q = 4; q < 9; ++q) lf *= ((lane >> (8 - q)) & 1) ? sq[q] : cq[q];

  float* mb = &Ms[wave * NSTATE];
  float mc[16];
#pragma unroll
  for (int t = 0; t < 16; ++t) {
    float p = lf;
    p *= ((t >> 3) & 1) ? sq[0] : cq[0];
    p *= ((t >> 2) & 1) ? sq[1] : cq[1];
    p *= ((t >> 1) & 1) ? sq[2] : cq[2];
    p *= ( t       & 1) ? sq[3] : cq[3];
    mc[t] = p;
    mb[t * 32 + lane] = p;               // lane-consecutive: conflict-free
  }

#if USE_ASYNC_LDS
  __builtin_amdgcn_s_wait_asynccnt(0);
#endif
  __syncthreads();

  // exp_i = sum_k Qk[i][k] * m[k] * m[k ^ u_i]
  float acc[NQ];
#pragma unroll
  for (int i = 0; i < NQ; ++i) acc[i] = 0.f;
#pragma unroll
  for (int i = 0; i < NQ; ++i) {
    const int u = Us[i];
#pragma unroll
    for (int t = 0; t < 16; ++t) {
      const int k = t * 32 + lane;
      acc[i] += Qs[i * NSTATE + k] * mc[t] * mb[k ^ u];  // (k^u): lane-permuted, conflict-free
    }
  }

  // wave32 butterfly reduction
#pragma unroll
  for (int i = 0; i < NQ; ++i) {
#pragma unroll
    for (int off = 16; off > 0; off >>= 1)
      acc[i] += __shfl_xor(acc[i], off, 32);
  }

  if (lane == 0) {
    const int obase = b * (NQ * 784) + hw;   // out[b, i, h, w]
#pragma unroll
    for (int i = 0; i < NQ; ++i) out[obase + i * 784] = acc[i];
  }
}

extern "C" void kernel_launch(void* const* d_in, const int* in_sizes, int n_in,
                              void* d_out, int out_size, void* d_ws, size_t ws_size,
                              hipStream_t stream) {
  (void)in_sizes; (void)n_in; (void)out_size; (void)d_ws; (void)ws_size;
  const float* x = (const float*)d_in[0];
  const float* w = (const float*)d_in[1];
  quanv_prep<<<1, 512, 0, stream>>>(w);
  quanv_main<<<12544 / ROWS_PER_BLOCK, 256, 0, stream>>>(x, (float*)d_out);
}